// HGNN_layer_35579509080183
// MI455X (gfx1250) — compile-verified
//
#include <hip/hip_runtime.h>
#include <hip/hip_bf16.h>

typedef __attribute__((ext_vector_type(2))) float v2f;
typedef __attribute__((ext_vector_type(8))) float v8f;

// ---------------------------------------------------------------------------
// Stage 0: inter_nw scalar from weight3 [128x128].
// tv = weight3[0]; dot = weight3 @ tv; cos_i = dot_i/(||tv||*||row_i||);
// inter_nw = mean(cos). One block of 128 threads, one row per thread.
// ---------------------------------------------------------------------------
__global__ __launch_bounds__(128)
void inter_nw_kernel(const float* __restrict__ w3, float* __restrict__ out)
{
    __shared__ float sdata[128];
    const int i = threadIdx.x;            // row index
    float dot = 0.f, nrm2 = 0.f, tvn2 = 0.f;
    #pragma unroll 4
    for (int j = 0; j < 128; ++j) {
        const float wij = w3[i * 128 + j];
        const float tvj = w3[j];          // row 0 == tv
        dot  += wij * tvj;
        nrm2 += wij * wij;
        tvn2 += tvj * tvj;
    }
    sdata[i] = dot / (sqrtf(tvn2) * sqrtf(nrm2));
    __syncthreads();
    for (int s = 64; s > 0; s >>= 1) {
        if (i < s) sdata[i] += sdata[i + s];
        __syncthreads();
    }
    if (i == 0) out[0] = sdata[0] * (1.0f / 128.0f);
}

// ---------------------------------------------------------------------------
// GEMM: C[M x 128] = (A[M x 128] @ W[128 x 128]) * scale
// One wave -> 16-row x 128-col strip using V_WMMA_F32_16X16X4_F32, K loop of 32.
// A fragment (16x4 f32): lanes 0-15 hold row m0+lane, K = k+0,k+1;
//                        lanes 16-31 hold row m0+lane-16, K = k+2,k+3.
// B fragment (4x16 f32): lanes 0-15 hold col n0+lane, rows k+0,k+1;
//                        lanes 16-31 hold col n0+lane-16, rows k+2,k+3.
// C/D (16x16 f32): vgpr r, lanes 0-15 -> (m0+r, n0+lane);
//                           lanes 16-31 -> (m0+8+r, n0+lane-16).
// Block = 256 threads = 8 waves = 128 rows. M must be a multiple of 16.
// ---------------------------------------------------------------------------
__global__ __launch_bounds__(256)
void gemm128_wmma_f32(const float* __restrict__ A,
                      const float* __restrict__ W,
                      float* __restrict__ C,
                      int M,
                      const float* __restrict__ scale_ptr)
{
    const int lane = threadIdx.x & 31;
    const int wave = threadIdx.x >> 5;
    const int m0   = (blockIdx.x * 8 + wave) * 16;
    if (m0 >= M) return;                  // wave-uniform: EXEC stays all-1s

    const float scale = scale_ptr ? scale_ptr[0] : 1.0f;

    const int   half   = lane >> 4;        // 0 or 1
    const int   l15    = lane & 15;
    const float* aBase = A + (size_t)(m0 + l15) * 128 + half * 2;
    const float* bBase = W + (size_t)(half * 2) * 128 + l15;

    v8f acc[8] = {};

    #pragma unroll 4
    for (int k = 0; k < 128; k += 4) {
        v2f a;
        a.x = aBase[k];
        a.y = aBase[k + 1];
        const float* wk = bBase + (size_t)k * 128;
        #pragma unroll
        for (int t = 0; t < 8; ++t) {
            v2f b;
            b.x = wk[t * 16];              // row k+half*2,   col n0+l15
            b.y = wk[t * 16 + 128];        // row k+half*2+1, col n0+l15
            acc[t] = __builtin_amdgcn_wmma_f32_16x16x4_f32(
                false, a, false, b, (short)0, acc[t], false, false);
        }
    }

    const int cRowBase = m0 + half * 8;
    #pragma unroll
    for (int t = 0; t < 8; ++t) {
        const int n = t * 16 + l15;
        #pragma unroll
        for (int r = 0; r < 8; ++r) {
            C[(size_t)(cRowBase + r) * 128 + n] = acc[t][r] * scale;
        }
    }
}

// ---------------------------------------------------------------------------
// Stage 2: edge[e][c] = relu( mean_{a: seq[e][a]>0} xw[seq[e][a]][c] )
// (softmax over 0/-inf mask == uniform over valid; uniform over all 32 if none)
// 256 threads = 2 edges per block, thread = channel. Coalesced 512B row reads.
// ---------------------------------------------------------------------------
__global__ __launch_bounds__(256)
void edge_gather_kernel(const float* __restrict__ xw,
                        const int* __restrict__ seq,
                        float* __restrict__ edge, int E)
{
    const int c = threadIdx.x & 127;
    const int e = blockIdx.x * 2 + (threadIdx.x >> 7);
    if (e >= E) return;

    const int* srow = seq + (size_t)e * 32;
    float acc = 0.f;
    int   cnt = 0;
    #pragma unroll 4
    for (int a = 0; a < 32; ++a) {
        const int idx = srow[a];
        if (idx > 0) {
            acc += xw[(size_t)idx * 128 + c];
            ++cnt;
        }
    }
    float v;
    if (cnt > 0) {
        v = acc / (float)cnt;
    } else {                               // all logits equal -> uniform 1/32
        float s = 0.f;
        for (int a = 0; a < 32; ++a) s += xw[(size_t)srow[a] * 128 + c];
        v = s * (1.0f / 32.0f);
    }
    edge[(size_t)e * 128 + c] = fmaxf(v, 0.0f);   // fused relu
}

// ---------------------------------------------------------------------------
// Stage 4: node[n][c] = mean_{k: useq[n][k]>0} e1[useq[n][k]][c]
// ---------------------------------------------------------------------------
__global__ __launch_bounds__(256)
void node_gather_kernel(const float* __restrict__ e1,
                        const int* __restrict__ useq,
                        float* __restrict__ out, int N)
{
    const int c = threadIdx.x & 127;
    const int n = blockIdx.x * 2 + (threadIdx.x >> 7);
    if (n >= N) return;

    const int* urow = useq + (size_t)n * 16;
    float acc = 0.f;
    int   cnt = 0;
    #pragma unroll 4
    for (int k = 0; k < 16; ++k) {
        const int idx = urow[k];
        if (idx > 0) {
            acc += e1[(size_t)idx * 128 + c];
            ++cnt;
        }
    }
    float v;
    if (cnt > 0) {
        v = acc / (float)cnt;
    } else {
        float s = 0.f;
        for (int k = 0; k < 16; ++k) s += e1[(size_t)urow[k] * 128 + c];
        v = s * (1.0f / 16.0f);
    }
    out[(size_t)n * 128 + c] = v;
}

// ---------------------------------------------------------------------------
// Host launcher
// ---------------------------------------------------------------------------
extern "C" void kernel_launch(void* const* d_in, const int* in_sizes, int n_in,
                              void* d_out, int out_size, void* d_ws, size_t ws_size,
                              hipStream_t stream)
{
    const float* x   = (const float*)d_in[0];   // [N,128]
    const int*   seq = (const int*)  d_in[1];   // [E,32]
    const int*   usq = (const int*)  d_in[2];   // [N,16]
    // d_in[3] = TextVector (unused: reference overwrites it with weight3[0])
    const float* w1  = (const float*)d_in[4];   // [128,128]
    const float* w2  = (const float*)d_in[5];   // [128,128]
    const float* w3  = (const float*)d_in[6];   // [128,128]

    const int N = in_sizes[0] / 128;            // 100000
    const int E = in_sizes[1] / 32;             // 50000

    // Workspace layout (bytes): [ xw : N*128*4 ][ edge : E*128*4 ][ e1 : E*128*4 ][ scalar ]
    char* ws = (char*)d_ws;
    const size_t xw_bytes   = (size_t)N * 128 * sizeof(float);
    const size_t edge_bytes = (size_t)E * 128 * sizeof(float);
    float* xw       = (float*)(ws);
    float* edge     = (float*)(ws + xw_bytes);
    float* e1       = (float*)(ws + xw_bytes + edge_bytes);
    float* inter_nw = (float*)(ws + xw_bytes + 2 * edge_bytes);

    // Stage 0: scalar inter_nw from weight3
    inter_nw_kernel<<<1, 128, 0, stream>>>(w3, inter_nw);

    // Stage 1: xw = (x @ w1) * inter_nw      (WMMA f32, N rows)
    gemm128_wmma_f32<<<(N + 127) / 128, 256, 0, stream>>>(x, w1, xw, N, inter_nw);

    // Stage 2: edge = relu(masked-mean gather of xw over seq)
    edge_gather_kernel<<<(E + 1) / 2, 256, 0, stream>>>(xw, seq, edge, E);

    // Stage 3: e1 = edge @ w2                (WMMA f32, E rows)
    gemm128_wmma_f32<<<(E + 127) / 128, 256, 0, stream>>>(edge, w2, e1, E, nullptr);

    // Stage 4: node = masked-mean gather of e1 over useq -> d_out
    node_gather_kernel<<<(N + 1) / 2, 256, 0, stream>>>(e1, usq, (float*)d_out, N);
}